// SurfaceArbitrageFreeLoss_9156870275553
// MI455X (gfx1250) — compile-verified
//
#include <hip/hip_runtime.h>
#include <hip/hip_bf16.h>
#include <stdint.h>

// ---------------------------------------------------------------------------
// SurfaceArbitrageFreeLoss for MI455X (gfx1250, wave32)
//   Stage 1: per-point MLP + analytic grads, per-block partial losses.
//            Weights staged to LDS via CDNA5 async global->LDS loads.
//            tanh via native v_tanh_f32.
//   Stage 2: deterministic partial-sum reduction using V_WMMA_F32_16X16X4_F32
//            with an all-ones B matrix (64 values summed per wmma).
// ---------------------------------------------------------------------------

#define SAF_EPS 1e-5f

typedef float v2f __attribute__((ext_vector_type(2)));
typedef float v8f __attribute__((ext_vector_type(8)));

constexpr int H   = 128;  // hidden width (reference fixes H=128)
constexpr int TPB = 256;  // 8 waves of 32
constexpr int PPT = 2;    // points per thread (ILP across v_tanh pipeline)

__device__ __forceinline__ float fast_tanhf(float x) {
#if __has_builtin(__builtin_amdgcn_tanhf)
  return __builtin_amdgcn_tanhf(x);
#else
  float r;
  asm volatile("v_tanh_f32 %0, %1\n\t"
               "s_delay_alu instid0(TRANS32_DEP_1)"
               : "=v"(r) : "v"(x));
  return r;
#endif
}

__device__ __forceinline__ void point_loss(float S0, float S1, float S2, float S3,
                                           float x, float tgt, float b2v,
                                           float& l0, float& l1, float& l2) {
  float s  = S0 + b2v;
  // softplus, stable
  float w  = (s > 15.f) ? s : __logf(1.f + __expf(s));
  float sg = __builtin_amdgcn_rcpf(1.f + __expf(-s));   // sigmoid(s)
  float w_t  = sg * S1;
  float w_x  = sg * S2;
  float w_xx = sg * (1.f - sg) * S2 * S2 - 2.f * sg * S3;

  float pe = (w - tgt) * __builtin_amdgcn_rcpf(tgt + SAF_EPS);
  l0 += pe * pe;

  float cal = fmaxf(-w_t, 0.f);
  l1 += cal * cal;

  float rw   = __builtin_amdgcn_rcpf(w);
  float term = fmaf(-0.5f * x * w_x, rw, 1.f);          // 1 - x*w_x/(2w)
  float g    = term * term - 0.25f * w_x * (rw + 0.25f) + 0.5f * w_xx;
  float bf   = fmaxf(-g, 0.f);
  l2 += bf * bf;
}

__global__ __launch_bounds__(TPB) void saf_main_kernel(
    const float* __restrict__ t_in, const float* __restrict__ x_in,
    const float* __restrict__ tgt_in,
    const float* __restrict__ W1, const float* __restrict__ b1,
    const float* __restrict__ W2, const float* __restrict__ b2,
    float* __restrict__ partials, int N, int nblk) {
  __shared__ float4 wlds[H];              // {a_i, c_i, b1_i, v_i} per unit
  __shared__ float  wred[3][TPB / 32];

  const int tid = threadIdx.x;

  // ---- Stage weights into LDS with async global->LDS gather (ASYNCcnt) ----
  for (int f = tid; f < 4 * H; f += TPB) {
    const int unit = f >> 2;
    const int slot = f & 3;
    const float* src = (slot == 0) ? (W1 + unit)
                     : (slot == 1) ? (W1 + H + unit)
                     : (slot == 2) ? (b1 + unit)
                                   : (W2 + unit);
    uint32_t ldsoff = (uint32_t)(uintptr_t)(((float*)wlds) + f);
    asm volatile("global_load_async_to_lds_b32 %0, %1, off"
                 :: "v"(ldsoff), "v"(src) : "memory");
  }
#if __has_builtin(__builtin_amdgcn_s_wait_asynccnt)
  __builtin_amdgcn_s_wait_asynccnt(0);
#else
  asm volatile("s_wait_asynccnt 0" ::: "memory");
#endif
  __syncthreads();

  // ---- Per-point MLP + analytic derivatives (2 points per thread) ----
  const int  i0 = blockIdx.x * (TPB * PPT) + tid;
  const int  i1 = i0 + TPB;
  const bool v0 = (i0 < N), v1 = (i1 < N);

  float t0 = 0.f, x0 = 0.f, t1 = 0.f, x1 = 0.f;
  if (v0) { t0 = t_in[i0]; x0 = x_in[i0]; }
  if (v1) { t1 = t_in[i1]; x1 = x_in[i1]; }
  const float b2v = b2[0];

  float S00 = 0.f, S01 = 0.f, S02 = 0.f, S03 = 0.f;
  float S10 = 0.f, S11 = 0.f, S12 = 0.f, S13 = 0.f;

#pragma unroll 4
  for (int i = 0; i < H; ++i) {
    const float4 wv = wlds[i];
    const float a = wv.x, c = wv.y, bb = wv.z, v = wv.w;
    const float cc = c * c;
    {
      float z  = fmaf(t0, a, fmaf(x0, c, bb));
      float h  = fast_tanhf(z);
      float q  = fmaf(-h, h, 1.f);          // 1 - h^2
      float vq = v * q;
      S00 = fmaf(v, h, S00);
      S01 = fmaf(vq, a, S01);
      S02 = fmaf(vq, c, S02);
      S03 = fmaf(vq * h, cc, S03);
    }
    {
      float z  = fmaf(t1, a, fmaf(x1, c, bb));
      float h  = fast_tanhf(z);
      float q  = fmaf(-h, h, 1.f);
      float vq = v * q;
      S10 = fmaf(v, h, S10);
      S11 = fmaf(vq, a, S11);
      S12 = fmaf(vq, c, S12);
      S13 = fmaf(vq * h, cc, S13);
    }
  }

  float l0 = 0.f, l1 = 0.f, l2 = 0.f;
  if (v0) point_loss(S00, S01, S02, S03, x0, tgt_in[i0], b2v, l0, l1, l2);
  if (v1) point_loss(S10, S11, S12, S13, x1, tgt_in[i1], b2v, l0, l1, l2);

  // ---- Deterministic block reduction (wave32 shuffles, then LDS) ----
  for (int off = 16; off > 0; off >>= 1) {
    l0 += __shfl_down(l0, off);
    l1 += __shfl_down(l1, off);
    l2 += __shfl_down(l2, off);
  }
  const int lane = tid & 31, wid = tid >> 5;
  if (lane == 0) { wred[0][wid] = l0; wred[1][wid] = l1; wred[2][wid] = l2; }
  __syncthreads();
  if (tid == 0) {
    float a0 = 0.f, a1 = 0.f, a2 = 0.f;
    for (int i = 0; i < TPB / 32; ++i) {
      a0 += wred[0][i]; a1 += wred[1][i]; a2 += wred[2][i];
    }
    partials[0 * nblk + blockIdx.x] = a0;
    partials[1 * nblk + blockIdx.x] = a1;
    partials[2 * nblk + blockIdx.x] = a2;
  }
}

// Stage 2: one wave reduces all block partials with V_WMMA_F32_16X16X4_F32.
// B = all-ones, so D[m,n] += sum_k A[m,k]; any bijective packing of 64
// partials into A sums correctly regardless of exact VGPR striping.
__global__ __launch_bounds__(32) void saf_reduce_kernel(
    const float* __restrict__ p, int nblk, int npad, float invN,
    float* __restrict__ out) {
  const int lane = threadIdx.x;
#if __has_builtin(__builtin_amdgcn_wmma_f32_16x16x4_f32)
  // A(16x4) f32 layout: lanes 0-15 hold K={0,1} of row M=lane,
  // lanes 16-31 hold K={2,3}. Flat chunk index pair: base, base+1.
  const int base = (lane & 15) * 4 + (lane >> 4) * 2;
  v2f bones; bones[0] = 1.f; bones[1] = 1.f;
  for (int j = 0; j < 3; ++j) {
    const float* pj = p + j * nblk;
    v8f c = {0.f, 0.f, 0.f, 0.f, 0.f, 0.f, 0.f, 0.f};
    for (int it = 0; it < npad; it += 64) {
      const int k = it + base;
      v2f a;
      a[0] = (k     < nblk) ? pj[k]     : 0.f;
      a[1] = (k + 1 < nblk) ? pj[k + 1] : 0.f;
      c = __builtin_amdgcn_wmma_f32_16x16x4_f32(
              /*neg_a=*/false, a, /*neg_b=*/false, bones,
              /*c_mod=*/(short)0, c, /*reuse_a=*/false, /*reuse_b=*/false);
    }
    float s = c[0] + c[1] + c[2] + c[3] + c[4] + c[5] + c[6] + c[7];
    s += __shfl_xor(s, 16);  // combine M=0..7 half with M=8..15 half
    if (lane == 0) out[j] = s * invN;
  }
#else
  if (lane == 0) {
    for (int j = 0; j < 3; ++j) {
      float s = 0.f;
      for (int i = 0; i < nblk; ++i) s += p[j * nblk + i];
      out[j] = s * invN;
    }
  }
  (void)npad;
#endif
}

extern "C" void kernel_launch(void* const* d_in, const int* in_sizes, int n_in,
                              void* d_out, int out_size, void* d_ws, size_t ws_size,
                              hipStream_t stream) {
  const float* t   = (const float*)d_in[0];
  const float* x   = (const float*)d_in[1];
  const float* tgt = (const float*)d_in[2];
  const float* W1  = (const float*)d_in[3];
  const float* b1  = (const float*)d_in[4];
  const float* W2  = (const float*)d_in[5];
  const float* b2  = (const float*)d_in[6];
  float* out       = (float*)d_out;
  float* partials  = (float*)d_ws;

  const int N    = in_sizes[0];
  const int nblk = (N + TPB * PPT - 1) / (TPB * PPT);
  const int npad = ((nblk + 63) / 64) * 64;

  saf_main_kernel<<<nblk, TPB, 0, stream>>>(t, x, tgt, W1, b1, W2, b2,
                                            partials, N, nblk);
  saf_reduce_kernel<<<1, 32, 0, stream>>>(partials, nblk, npad,
                                          1.0f / (float)N, out);
}